// TiDHy_2405181686601
// MI455X (gfx1250) — compile-verified
//
#include <hip/hip_runtime.h>

// ---------------------------------------------------------------------------
// TiDHy reference collapses algebraically:
//   r carry is always zeros -> r_bar == 0, temp_loss == 0, r2 stays zeros.
//   sl_rhat = sum_{t=0..31} mean_b sum_d (b_sd[d]-X[b,t,d])^2
//   sl_rbar = sl_rhat - (t==0 term)
// Remaining work: one streaming squared-error reduction over X (8 MB) plus
// zero-fill of r0/r2 outputs. Memory-bound: ~8.4 MB / 23.3 TB/s ≈ 0.36 us.
//
// Sum-of-squares accumulation runs through v_wmma_f32_16x16x4_f32 via the
// ones-matmul reduction (B = all ones -> sum(D entries) == 16 * sum(A
// entries), independent of A/B slot layout -> zero layout risk).
//
// Grid is fixed at 1024 waves over 4096 rows: each wave does exactly 4
// fully-unrolled iterations -> straight-line code, EXEC provably all-ones at
// every WMMA (ISA requirement), no exec-masked loop structurization.
// Rows of wave g are g, g+1024, g+2048, g+3072, all congruent mod 32, so the
// t==0 slice is exactly the waves with g % 32 == 0 -> no second accumulator.
// ---------------------------------------------------------------------------

typedef __attribute__((ext_vector_type(2))) float v2f;
typedef __attribute__((ext_vector_type(8))) float v8f;

#define BB   128
#define TT   32
#define DD   512
#define ROWS (BB * TT)      // 4096 (b,t) rows of length D=512
#define NPART 1024          // waves in stage-1 grid (128 blocks * 8 waves)
#define ITERS (ROWS / NPART) // 4 rows per wave, compile-time
#define ZERO_ELEMS 49152    // r0 (128*256) + r2 (128*128)

__device__ __forceinline__ float sumsq4(float4 x, float4 b) {
    float e0 = x.x - b.x, e1 = x.y - b.y, e2 = x.z - b.z, e3 = x.w - b.w;
    return e0 * e0 + e1 * e1 + e2 * e2 + e3 * e3;
}

// Stage 1: each wave streams 4 whole 512-float rows; WMMA accumulates sum(e^2).
__global__ __launch_bounds__(256) void tidhy_sq_reduce(
    const float* __restrict__ X, const float* __restrict__ b_sd,
    float* __restrict__ part_all, float* __restrict__ part_t0)
{
    const int lane  = threadIdx.x & 31;
    const int gwave = blockIdx.x * (blockDim.x >> 5) + (threadIdx.x >> 5);

    // b_sd is invariant per lane across all 4 rows: d = lane*4 + j*128
    const float4* b4 = (const float4*)b_sd;
    const float4 bv0 = b4[lane];
    const float4 bv1 = b4[lane + 32];
    const float4 bv2 = b4[lane + 64];
    const float4 bv3 = b4[lane + 96];

    const v2f ones = {1.0f, 1.0f};   // B matrix: layout-immaterial
    v8f c = {};                      // WMMA accumulator: holds 16 * sum(e^2)

    #pragma unroll
    for (int k = 0; k < ITERS; ++k) {
        const int row = gwave + k * NPART;
        const float4* xr = (const float4*)(X + (size_t)row * DD) + lane;
        if (k + 1 < ITERS)           // compile-time guard -> no divergence
            __builtin_prefetch(X + (size_t)(row + NPART) * DD + lane * 4, 0, 0);

        float4 x0 = xr[0];
        float4 x1 = xr[32];
        float4 x2 = xr[64];
        float4 x3 = xr[96];

        v2f a;
        a.x = sumsq4(x0, bv0);
        a.y = sumsq4(x1, bv1);
        c = __builtin_amdgcn_wmma_f32_16x16x4_f32(
                false, a, false, ones, (short)0, c, false, false);
        a.x = sumsq4(x2, bv2);
        a.y = sumsq4(x3, bv3);
        c = __builtin_amdgcn_wmma_f32_16x16x4_f32(
                false, a, false, ones, (short)0, c, false, false);
    }

    // Collapse accumulator: sum of all 256 D entries == 16 * wave's sum(e^2).
    float ls = c[0] + c[1] + c[2] + c[3] + c[4] + c[5] + c[6] + c[7];
    #pragma unroll
    for (int m = 16; m > 0; m >>= 1)
        ls += __shfl_xor(ls, m, 32);

    if (lane == 0) {
        const float s = ls * (1.0f / 16.0f);
        part_all[gwave] = s;
        // t = row % 32 is constant per wave; t==0 slice <=> gwave % 32 == 0
        part_t0[gwave]  = ((gwave & (TT - 1)) == 0) ? s : 0.0f;
    }
}

// Stage 2: deterministic fixed-order reduction of the 1024 wave partials.
__global__ __launch_bounds__(512) void tidhy_finalize(
    const float* __restrict__ part_all, const float* __restrict__ part_t0,
    float* __restrict__ out)
{
    __shared__ float sA[512];
    __shared__ float sZ[512];
    const int t = threadIdx.x;
    sA[t] = part_all[t] + part_all[t + 512];
    sZ[t] = part_t0[t]  + part_t0[t + 512];
    __syncthreads();
    #pragma unroll
    for (int s = 256; s > 0; s >>= 1) {
        if (t < s) { sA[t] += sA[t + s]; sZ[t] += sZ[t + s]; }
        __syncthreads();
    }
    if (t == 0) {
        const float S_all = sA[0];   // sum_{b,t,d} e^2
        const float S_t0  = sZ[0];   // t==0 slice
        out[0] = S_all * (1.0f / (float)BB);            // sl_rhat
        out[1] = (S_all - S_t0) * (1.0f / (float)BB);   // sl_rbar
        out[2] = 0.0f;                                  // TEMP_WEIGHT * temp_loss
        out[3] = 0.0f;                                  // r2_losses
    }
}

// Zero-fill r0 (128x256) and r2 (128x128) outputs.
__global__ __launch_bounds__(256) void tidhy_zero(float* __restrict__ out)
{
    const int i = blockIdx.x * blockDim.x + threadIdx.x;   // exact: 192*256
    out[4 + i] = 0.0f;
}

extern "C" void kernel_launch(void* const* d_in, const int* in_sizes, int n_in,
                              void* d_out, int out_size, void* d_ws, size_t ws_size,
                              hipStream_t stream)
{
    (void)in_sizes; (void)n_in; (void)out_size; (void)ws_size;
    const float* X    = (const float*)d_in[0];   // (128, 32, 512)
    const float* b_sd = (const float*)d_in[3];   // (512,)
    float* out = (float*)d_out;                  // 49156 f32

    float* part_all = (float*)d_ws;              // NPART floats
    float* part_t0  = part_all + NPART;          // NPART floats

    tidhy_zero<<<ZERO_ELEMS / 256, 256, 0, stream>>>(out);
    tidhy_sq_reduce<<<NPART / 8, 256, 0, stream>>>(X, b_sd, part_all, part_t0);
    tidhy_finalize<<<1, 512, 0, stream>>>(part_all, part_t0, out);
}